// AbstractQCP_60533269070251
// MI455X (gfx1250) — compile-verified
//
#include <hip/hip_runtime.h>

// ---------------------------------------------------------------------------
// AbstractQCP differentiation via CGLS (60 iters), fp32, L2-resident (~60 MB
// working set vs 192 MB L2). SpMV scatter-add uses native global_atomic_add_f32
// (unsafeAtomicAdd) with 4-nnz-per-thread B128 streaming loads. Dot-product
// wave reductions use V_WMMA_F32_16X16X4_F32 (ones-matrix trick).
// ---------------------------------------------------------------------------

typedef __attribute__((ext_vector_type(2))) float v2f;
typedef __attribute__((ext_vector_type(8))) float v8f;

#define EPS 1e-30f

// Sum 'acc' across all 32 lanes of the wave using one f32 WMMA.
// A[16x4]: lane L<16 -> A[L][0]=acc ; lane L>=16 -> A[L-16][2]=acc (other K = 0).
// B[4x16] = all ones  =>  D[i][j] = acc[i] + acc[i+16] for every column j.
// Each lane sums its 8 D rows; pairing lane L with L^16 yields the full sum in all lanes.
__device__ __forceinline__ float aqcp_wave_sum(float acc) {
    v2f a; a.x = acc; a.y = 0.0f;
    v2f bo; bo.x = 1.0f; bo.y = 1.0f;
    v8f c = {};
    c = __builtin_amdgcn_wmma_f32_16x16x4_f32(
            /*neg_a=*/false, a, /*neg_b=*/false, bo,
            /*c_mod=*/(short)0, c, /*reuse_a=*/false, /*reuse_b=*/false);
    float p = c[0] + c[1] + c[2] + c[3] + c[4] + c[5] + c[6] + c[7];
    p += __shfl_xor(p, 16, 32);
    return p;
}

// out[0] = sum_i a[i]*b[i]*(m3 ? m3[i] : 1).  Single workgroup of 1024 (32 waves).
__global__ __launch_bounds__(1024) void aqcp_dot(const float* __restrict__ a,
                                                 const float* __restrict__ b,
                                                 const float* __restrict__ m3,
                                                 int len, float* __restrict__ out) {
    __shared__ float lds[32];
    int tid = threadIdx.x;
    float acc = 0.0f;
    int len4 = len & ~3;
    if (m3) {
        for (int i = tid * 4; i < len4; i += 4096) {
            float4 av = *(const float4*)(a + i);
            float4 bv = *(const float4*)(b + i);
            float4 mv = *(const float4*)(m3 + i);
            acc += av.x * bv.x * mv.x + av.y * bv.y * mv.y +
                   av.z * bv.z * mv.z + av.w * bv.w * mv.w;
        }
        for (int i = len4 + tid; i < len; i += 1024) acc += a[i] * b[i] * m3[i];
    } else {
        for (int i = tid * 4; i < len4; i += 4096) {
            float4 av = *(const float4*)(a + i);
            float4 bv = *(const float4*)(b + i);
            acc += av.x * bv.x + av.y * bv.y + av.z * bv.z + av.w * bv.w;
        }
        for (int i = len4 + tid; i < len; i += 1024) acc += a[i] * b[i];
    }
    float w = aqcp_wave_sum(acc);
    int wave = tid >> 5, lane = tid & 31;
    if (lane == 0) lds[wave] = w;
    __syncthreads();
    if (wave == 0) {
        float t = aqcp_wave_sum(lds[lane]);
        if (lane == 0) out[0] = t;
    }
}

// out[sidx[i]] += scale*data[i]*vec[gidx[i]]*(gmask?gmask[gidx[i]]:1)*(smask?smask[sidx[i]]:1)
// 4 nnz per thread with B128 streaming loads; scalar tail.
__global__ void aqcp_spmv4(const float* __restrict__ data, const int* __restrict__ gidx,
                           const int* __restrict__ sidx, const float* __restrict__ vec,
                           const float* __restrict__ gmask, const float* __restrict__ smask,
                           float scale, float* __restrict__ out, int nnz) {
    int i = (blockIdx.x * blockDim.x + threadIdx.x) * 4;
    if (i + 3 < nnz) {
        float4 dv = *(const float4*)(data + i);
        int4   gv = *(const int4*)(gidx + i);
        int4   sv = *(const int4*)(sidx + i);
        float v0 = dv.x * vec[gv.x] * scale;
        float v1 = dv.y * vec[gv.y] * scale;
        float v2 = dv.z * vec[gv.z] * scale;
        float v3 = dv.w * vec[gv.w] * scale;
        if (gmask) {
            v0 *= gmask[gv.x]; v1 *= gmask[gv.y];
            v2 *= gmask[gv.z]; v3 *= gmask[gv.w];
        }
        if (smask) {
            v0 *= smask[sv.x]; v1 *= smask[sv.y];
            v2 *= smask[sv.z]; v3 *= smask[sv.w];
        }
        unsafeAtomicAdd(&out[sv.x], v0);
        unsafeAtomicAdd(&out[sv.y], v1);
        unsafeAtomicAdd(&out[sv.z], v2);
        unsafeAtomicAdd(&out[sv.w], v3);
    } else {
        for (int k = i; k < nnz; ++k) {
            int g  = gidx[k];
            int sI = sidx[k];
            float v = data[k] * vec[g] * scale;
            if (gmask) v *= gmask[g];
            if (smask) v *= smask[sI];
            unsafeAtomicAdd(&out[sI], v);
        }
    }
}

// mask = (y-s > 0), pim = max(y-s, 0)
__global__ void aqcp_maskpim(const float* __restrict__ y, const float* __restrict__ s,
                             float* __restrict__ maskv, float* __restrict__ pim, int m) {
    int j = blockIdx.x * blockDim.x + threadIdx.x;
    if (j >= m) return;
    float v = y[j] - s[j];
    maskv[j] = (v > 0.0f) ? 1.0f : 0.0f;
    pim[j]   = fmaxf(v, 0.0f);
}

__global__ void aqcp_init_qg_tmpn(float* __restrict__ qg, float* __restrict__ tmpn,
                                  const float* __restrict__ q, int n) {
    int i = blockIdx.x * blockDim.x + threadIdx.x;
    if (i >= n) return;
    qg[i] = q[i];
    tmpn[i] = 0.0f;
}

__global__ void aqcp_vadd(float* __restrict__ dst, const float* __restrict__ src, int n) {
    int i = blockIdx.x * blockDim.x + threadIdx.x;
    if (i < n) dst[i] += src[i];
}

__global__ void aqcp_zero(float* __restrict__ dst, int n) {
    int i = blockIdx.x * blockDim.x + threadIdx.x;
    if (i < n) dst[i] = 0.0f;
}

// z = 0 ; r = [-dq, -db, 0]  (r := rhs = -d, last entry filled later)
__global__ void aqcp_init_rhs_z(float* __restrict__ z, float* __restrict__ r,
                                const float* __restrict__ dq, const float* __restrict__ db,
                                int n, int m, int N) {
    int i = blockIdx.x * blockDim.x + threadIdx.x;
    if (i >= N) return;
    z[i] = 0.0f;
    if (i < n)            r[i] = -dq[i];
    else if (i < n + m)   r[i] = -db[i - n];
    else                  r[i] = 0.0f;
}

// F(u) init: out1 = q*u3 ; out2 = b*u3 + (1-mask)*u2 ; out3 = 0 (set later)
__global__ void aqcp_initF(float* __restrict__ outv, const float* __restrict__ u,
                           const float* __restrict__ q, const float* __restrict__ b,
                           const float* __restrict__ maskv, int n, int m, int N) {
    int i = blockIdx.x * blockDim.x + threadIdx.x;
    if (i >= N) return;
    float u3 = u[N - 1];
    if (i < n)          outv[i] = q[i] * u3;
    else if (i < n + m) { int j = i - n; outv[i] = b[j] * u3 + (1.0f - maskv[j]) * u[i]; }
    else                outv[i] = 0.0f;
}

// FT(u) init: out1 = -qg*u3 ; out2 = mask*(-b*u3) + (1-mask)*u2 ; out3 = 0
__global__ void aqcp_initFT(float* __restrict__ outv, const float* __restrict__ u,
                            const float* __restrict__ qg, const float* __restrict__ b,
                            const float* __restrict__ maskv, int n, int m, int N) {
    int i = blockIdx.x * blockDim.x + threadIdx.x;
    if (i >= N) return;
    float u3 = u[N - 1];
    if (i < n)          outv[i] = -qg[i] * u3;
    else if (i < n + m) { int j = i - n; float mk = maskv[j];
                          outv[i] = mk * (-b[j] * u3) + (1.0f - mk) * u[i]; }
    else                outv[i] = 0.0f;
}

// Scalars: S[0]=c  S[1]=gk  S[2]=alpha  S[3]=beta  S[4],S[5],S[6]=dot temps  S[8]=qq/gn
__global__ void aqcp_rhs3(float* __restrict__ r, const float* __restrict__ S, int N) {
    r[N - 1] = S[4] + S[5] + S[6];
}
__global__ void aqcp_r3F(float* __restrict__ qv, const float* __restrict__ p,
                         const float* __restrict__ S, int N) {
    qv[N - 1] = -S[4] - S[5] + S[0] * p[N - 1];
}
__global__ void aqcp_r3FT(float* __restrict__ sv, const float* __restrict__ r,
                          const float* __restrict__ S, int N) {
    sv[N - 1] = S[4] + S[5] + S[0] * r[N - 1];
}
__global__ void aqcp_alpha(float* __restrict__ S) {
    S[2] = S[1] / (S[8] + EPS);
}
__global__ void aqcp_beta(float* __restrict__ S) {
    S[3] = S[8] / (S[1] + EPS);
    S[1] = S[8];
}

// z += alpha*p ; r -= alpha*q
__global__ void aqcp_axpy2(float* __restrict__ z, float* __restrict__ r,
                           const float* __restrict__ p, const float* __restrict__ qv,
                           const float* __restrict__ S, int N) {
    int i = blockIdx.x * blockDim.x + threadIdx.x;
    if (i >= N) return;
    float al = S[2];
    z[i] += al * p[i];
    r[i] -= al * qv[i];
}

// p = sv + beta*p
__global__ void aqcp_xpay(float* __restrict__ p, const float* __restrict__ sv,
                          const float* __restrict__ S, int N) {
    int i = blockIdx.x * blockDim.x + threadIdx.x;
    if (i >= N) return;
    p[i] = sv[i] + S[3] * p[i];
}

// dx = z1 - x*zN ; t = mask*z2 ; dy = t - y*zN ; ds = t - z2 - s*zN
__global__ void aqcp_out(float* __restrict__ outp, const float* __restrict__ z,
                         const float* __restrict__ x, const float* __restrict__ y,
                         const float* __restrict__ s, const float* __restrict__ maskv,
                         int n, int m, int N) {
    int i = blockIdx.x * blockDim.x + threadIdx.x;
    if (i >= n + m) return;
    float zN = z[N - 1];
    if (i < n) {
        outp[i] = z[i] - x[i] * zN;
    } else {
        int j = i - n;
        float z2 = z[n + j];
        float t = maskv[j] * z2;
        outp[n + j]     = t - y[j] * zN;
        outp[n + m + j] = t - z2 - s[j] * zN;
    }
}

extern "C" void kernel_launch(void* const* d_in, const int* in_sizes, int n_in,
                              void* d_out, int out_size, void* d_ws, size_t ws_size,
                              hipStream_t stream) {
    (void)n_in; (void)out_size; (void)ws_size;
    const float* Pd  = (const float*)d_in[0];
    const int*   Pr  = (const int*)  d_in[1];
    const int*   Pc  = (const int*)  d_in[2];
    const float* Ad  = (const float*)d_in[3];
    const int*   Ar  = (const int*)  d_in[4];
    const int*   Ac  = (const int*)  d_in[5];
    const float* q   = (const float*)d_in[6];
    const float* b   = (const float*)d_in[7];
    const float* x   = (const float*)d_in[8];
    const float* y   = (const float*)d_in[9];
    const float* s   = (const float*)d_in[10];
    const float* dPd = (const float*)d_in[11];
    const float* dAd = (const float*)d_in[12];
    const float* dq  = (const float*)d_in[13];
    const float* db  = (const float*)d_in[14];
    float* outp = (float*)d_out;

    const int nnzP = in_sizes[0];
    const int nnzA = in_sizes[3];
    const int n = in_sizes[6];
    const int m = in_sizes[7];
    const int N = n + m + 1;

    // scratch layout (floats); total ~ 2n+2m+5N+16 ≈ 1.7 MB
    float* W     = (float*)d_ws;
    float* qg    = W;                  // n
    float* pim   = qg + n;             // m
    float* maskv = pim + m;            // m
    float* tmpn  = maskv + m;          // n
    float* z     = tmpn + n;           // N
    float* r     = z + N;              // N
    float* p     = r + N;              // N
    float* qv    = p + N;              // N
    float* sv    = qv + N;             // N
    float* S     = sv + N;             // 16 scalars

    const int TB = 256;
    auto gb = [](int t) { return (t + 255) / 256; };

    auto spmv = [&](const float* data, const int* gi, const int* si, const float* vec,
                    const float* gm, const float* sm, float sc, float* out, int nnz) {
        int threads = (nnz + 3) / 4;
        aqcp_spmv4<<<gb(threads), TB, 0, stream>>>(data, gi, si, vec, gm, sm, sc, out, nnz);
    };
    auto dot = [&](const float* a, const float* bb, const float* mm, int len, float* out) {
        aqcp_dot<<<1, 1024, 0, stream>>>(a, bb, mm, len, out);
    };

    // ---- setup: mask/pi_m, qg = q + Px + P^T x, c = x·Px -------------------
    aqcp_maskpim<<<gb(m), TB, 0, stream>>>(y, s, maskv, pim, m);
    aqcp_init_qg_tmpn<<<gb(n), TB, 0, stream>>>(qg, tmpn, q, n);
    spmv(Pd, Pc, Pr, x, nullptr, nullptr, 1.0f, tmpn, nnzP);      // tmpn = P x
    dot(x, tmpn, nullptr, n, &S[0]);                              // c = x·Px
    aqcp_vadd<<<gb(n), TB, 0, stream>>>(qg, tmpn, n);             // qg = q + Px
    spmv(Pd, Pr, Pc, x, nullptr, nullptr, 1.0f, qg, nnzP);        // qg += P^T x

    // ---- rhs = -d ----------------------------------------------------------
    aqcp_init_rhs_z<<<gb(N), TB, 0, stream>>>(z, r, dq, db, n, m, N);
    spmv(dPd, Pc, Pr, x,   nullptr, nullptr, -1.0f, r,     nnzP); // -dP x
    spmv(dAd, Ar, Ac, pim, nullptr, nullptr, -1.0f, r,     nnzA); // -dA^T pi_m
    spmv(dAd, Ac, Ar, x,   nullptr, nullptr,  1.0f, r + n, nnzA); // +dA x
    aqcp_zero<<<gb(n), TB, 0, stream>>>(tmpn, n);
    spmv(dPd, Pc, Pr, x, nullptr, nullptr, 1.0f, tmpn, nnzP);     // tmpn = dP x
    dot(x,  tmpn, nullptr, n, &S[6]);                             // x·(dP x)
    dot(dq, x,    nullptr, n, &S[4]);                             // dq·x
    dot(db, pim,  nullptr, m, &S[5]);                             // db·pi_m
    aqcp_rhs3<<<1, 1, 0, stream>>>(r, S, N);                      // r[N-1] = sum

    // ---- FT(u) -> dst ------------------------------------------------------
    auto FT_apply = [&](float* dst, const float* u) {
        aqcp_initFT<<<gb(N), TB, 0, stream>>>(dst, u, qg, b, maskv, n, m, N);
        spmv(Pd, Pr, Pc, u,     nullptr, nullptr,  1.0f, dst,     nnzP); // + P^T u1
        spmv(Ad, Ar, Ac, u + n, nullptr, nullptr, -1.0f, dst,     nnzA); // - A^T u2
        spmv(Ad, Ac, Ar, u,     nullptr, maskv,    1.0f, dst + n, nnzA); // + mask*(A u1)
        dot(q, u,     nullptr, n, &S[4]);
        dot(b, u + n, nullptr, m, &S[5]);
        aqcp_r3FT<<<1, 1, 0, stream>>>(dst, u, S, N);
    };
    // ---- F(u) -> dst -------------------------------------------------------
    auto F_apply = [&](float* dst, const float* u) {
        aqcp_initF<<<gb(N), TB, 0, stream>>>(dst, u, q, b, maskv, n, m, N);
        spmv(Pd, Pc, Pr, u,     nullptr, nullptr,  1.0f, dst,     nnzP); // + P u1
        spmv(Ad, Ar, Ac, u + n, maskv,   nullptr,  1.0f, dst,     nnzA); // + A^T(mask u2)
        spmv(Ad, Ac, Ar, u,     nullptr, nullptr, -1.0f, dst + n, nnzA); // - A u1
        dot(qg, u,     nullptr, n, &S[4]);
        dot(b,  u + n, maskv,   m, &S[5]);
        aqcp_r3F<<<1, 1, 0, stream>>>(dst, u, S, N);
    };

    // ---- CGLS init: p = FT(r) ; gk = p·p -----------------------------------
    FT_apply(p, r);
    dot(p, p, nullptr, N, &S[1]);

    // ---- 60 CGLS iterations ------------------------------------------------
    for (int it = 0; it < 60; ++it) {
        F_apply(qv, p);
        dot(qv, qv, nullptr, N, &S[8]);                       // qq
        aqcp_alpha<<<1, 1, 0, stream>>>(S);                   // alpha = gk/(qq+eps)
        aqcp_axpy2<<<gb(N), TB, 0, stream>>>(z, r, p, qv, S, N);
        FT_apply(sv, r);
        dot(sv, sv, nullptr, N, &S[8]);                       // gn
        aqcp_beta<<<1, 1, 0, stream>>>(S);                    // beta = gn/(gk+eps); gk = gn
        aqcp_xpay<<<gb(N), TB, 0, stream>>>(p, sv, S, N);
    }

    // ---- outputs -----------------------------------------------------------
    aqcp_out<<<gb(n + m), TB, 0, stream>>>(outp, z, x, y, s, maskv, n, m, N);
}